// HenonLayer_17540646437315
// MI455X (gfx1250) — compile-verified
//
#include <hip/hip_runtime.h>

// CDNA5 / gfx1250 Henon-map kernel.
// State per row: (X, Y). 4 iterations of:
//   Ylast = Y/2
//   gradV = sum_k (1 - tanh^2(Ylast*Win[k] + bin[k])) * Win[k]*Wout[k]/2
//   Xout = Y + eta ; Yout = -X + gradV
//
// 1 - tanh^2(y) = sech^2(y) = 4*t*r*r with t = exp2(2*log2e*y), r = rcp(t+1).
// The 32-wide weighted reduce is done on the matrix pipe:
//   A[m,k] = 2*Win[k]*Wout[k] (replicated over m), B[k,n] = sech^2/4 of row n
//   => D[m,n] = gradV[n]: batch index in the lane dim, every lane reads c[0].
// V_WMMA_F32_16X16X4_F32 layouts put (row = lane&15, k = 4b+2*half+{0,1})
// exactly where each lane computes its activations -> no shuffles needed.

typedef float v2f __attribute__((ext_vector_type(2)));
typedef float v8f __attribute__((ext_vector_type(8)));

#define ROWS_PER_WAVE 16
#define WAVES_PER_BLOCK 8
#define ROWS_PER_BLOCK (ROWS_PER_WAVE * WAVES_PER_BLOCK)

__global__ __launch_bounds__(256) void henon4_wmma_kernel(
    const float* __restrict__ z,      // (B,2) interleaved X,Y
    const float* __restrict__ Win,    // (32,)
    const float* __restrict__ Wout,   // (32,)
    const float* __restrict__ bin,    // (32,)
    const float* __restrict__ eta_p,  // (1,)
    float* __restrict__ out,          // (B,2)
    int nRows)
{
    const int tid  = threadIdx.x;
    const int lane = tid & 31;
    const int wave = tid >> 5;
    const int m    = lane & 15;   // batch row within wave tile (N index)
    const int half = lane >> 4;   // K sub-block selector
    int row = blockIdx.x * ROWS_PER_BLOCK + wave * ROWS_PER_WAVE + m;
    row = row < nRows ? row : nRows - 1;   // keep EXEC all-1s for WMMA

    // Per-lane K indices: k = 4*b + 2*half + {0,1}  (contiguous pair -> b64 loads)
    const int koff = 2 * half;

    v2f win2[8], bin2[8], wq2[8];
#pragma unroll
    for (int b = 0; b < 8; ++b) {
        const int k = 4 * b + koff;
        v2f wi = *(const v2f*)(Win + k);
        v2f wo = *(const v2f*)(Wout + k);
        v2f bi = *(const v2f*)(bin + k);
        win2[b] = wi;
        bin2[b] = bi;
        // A value: 4 * (Win*Wout/YDIAM) = 2*Win*Wout  (B side carries sech^2/4)
        wq2[b].x = 2.0f * wi.x * wo.x;
        wq2[b].y = 2.0f * wi.y * wo.y;
    }

    const float eta = *eta_p;
    v2f xy = *(const v2f*)(z + 2 * row);
    float X = xy.x;
    float Y = xy.y;

    const float K2 = 2.885390081777927f; // 2*log2(e)

#pragma unroll
    for (int it = 0; it < 4; ++it) {
        const float Ylast = Y * 0.5f;   // (Y - YMEAN)/YDIAM, YMEAN=0, YDIAM=2
        v8f c = {0.f, 0.f, 0.f, 0.f, 0.f, 0.f, 0.f, 0.f};
#pragma unroll
        for (int b = 0; b < 8; ++b) {
            // y = Ylast*Win[k] + bin[k] for this lane's two K values
            float y0 = fmaf(Ylast, win2[b].x, bin2[b].x);
            float y1 = fmaf(Ylast, win2[b].y, bin2[b].y);
            // sech^2(y)/4 = t*r*r, t = 2^(K2*y), r = 1/(t+1); clamp exponent
            float a0 = fminf(fmaxf(y0 * K2, -126.0f), 126.0f);
            float a1 = fminf(fmaxf(y1 * K2, -126.0f), 126.0f);
            float t0 = __builtin_amdgcn_exp2f(a0);
            float t1 = __builtin_amdgcn_exp2f(a1);
            float r0 = __builtin_amdgcn_rcpf(t0 + 1.0f);
            float r1 = __builtin_amdgcn_rcpf(t1 + 1.0f);
            v2f Bm;
            Bm.x = t0 * r0 * r0;
            Bm.y = t1 * r1 * r1;
            v2f Am;
            Am.x = wq2[b].x;
            Am.y = wq2[b].y;
            // D = A(16x4: weights bcast over M) * B(4x16: activations, N=batch)
            // + C ; accumulate K=32 over 8 chained WMMAs.
            c = __builtin_amdgcn_wmma_f32_16x16x4_f32(
                    false, Am, false, Bm, (short)0, c, false, false);
        }
        const float gradV = c[0];   // all 8 D rows identical; N = this lane's row
        const float Xn = Y + eta;
        const float Yn = gradV - X;
        X = Xn;
        Y = Yn;
    }

    v2f res;
    res.x = X;
    res.y = Y;
    *(v2f*)(out + 2 * row) = res;
}

extern "C" void kernel_launch(void* const* d_in, const int* in_sizes, int n_in,
                              void* d_out, int out_size, void* d_ws, size_t ws_size,
                              hipStream_t stream) {
    const float* z    = (const float*)d_in[0];   // (B,2)
    const float* Win  = (const float*)d_in[1];   // (1,32)
    const float* Wout = (const float*)d_in[2];   // (32,1)
    const float* bin  = (const float*)d_in[3];   // (1,32)
    const float* eta  = (const float*)d_in[4];   // (1,1)
    float* out = (float*)d_out;

    const int B = in_sizes[0] / 2;               // 1048576
    const int blocks = (B + ROWS_PER_BLOCK - 1) / ROWS_PER_BLOCK;  // 8192

    henon4_wmma_kernel<<<blocks, 256, 0, stream>>>(z, Win, Wout, bin, eta, out, B);
}